// BodyOrderedModel_NonLinear_15358803050682
// MI455X (gfx1250) — compile-verified
//
#include <hip/hip_runtime.h>
#include <math.h>

// BodyOrderedModel (MACE-like) forward + full analytic backward on gfx1250.
// Scatter-bound (~6 GB gather/scatter fwd+bwd -> ~300us @ 23.3 TB/s HBM);
// all channel GEMMs (fwd and transposed bwd) use exact-precision
// V_WMMA_F32_16X16X4_F32 (matrix math is nowhere near the bottleneck, so f32
// precision is free).  Workspace: ~154M floats (~617 MB).

typedef __attribute__((ext_vector_type(2))) float v2f;
typedef __attribute__((ext_vector_type(8))) float v8f;

namespace {

constexpr int N   = 20000;
constexpr int E   = 320000;
constexpr int C   = 64;
constexpr int NB  = 8;
constexpr int NE  = 4;
constexpr int G   = 16;
constexpr int L2  = 9;
constexpr int HID = 16;
constexpr int NL  = L2 * C;            // 576 floats per node state
constexpr float RMAX = 5.0f;
constexpr float PI_F = 3.14159265358979323846f;

__device__ __forceinline__ v8f wmma4(v2f a, v2f b, v8f c) {
  // D = A(16x4,f32) * B(4x16,f32) + C(16x16,f32); wave32, exact f32.
  return __builtin_amdgcn_wmma_f32_16x16x4_f32(false, a, false, b, (short)0, c,
                                               false, false);
}

__device__ __forceinline__ float siluf(float x) {
  return x / (1.0f + expf(-x));
}
__device__ __forceinline__ float dsiluf(float x) {
  float sg = 1.0f / (1.0f + expf(-x));
  return sg + x * sg * (1.0f - sg);
}

// ---------------------------------------------------------------- zero fill
__global__ void k_zero(float* __restrict__ p, long n) {
  long i = (long)blockIdx.x * blockDim.x + threadIdx.x;
  if (i < n) p[i] = 0.0f;
}

// --------------------------------- per-edge geometry: Y, radial basis, r, u
__global__ void k_geom(const float* __restrict__ pos,
                       const float* __restrict__ shifts,
                       const int* __restrict__ snd, const int* __restrict__ rcv,
                       float* __restrict__ Y, float* __restrict__ ef,
                       float* __restrict__ rbuf, float* __restrict__ ubuf) {
  int e = blockIdx.x * blockDim.x + threadIdx.x;
  if (e >= E) return;
  int s = snd[e], d = rcv[e];
  float vx = pos[3 * s]     - pos[3 * d]     + shifts[3 * e];
  float vy = pos[3 * s + 1] - pos[3 * d + 1] + shifts[3 * e + 1];
  float vz = pos[3 * s + 2] - pos[3 * d + 2] + shifts[3 * e + 2];
  float r  = sqrtf(vx * vx + vy * vy + vz * vz + 1e-12f);
  float inv = 1.0f / r;
  float x = vx * inv, y = vy * inv, z = vz * inv;
  rbuf[e] = r;
  ubuf[3 * e] = x; ubuf[3 * e + 1] = y; ubuf[3 * e + 2] = z;
  const float s3 = 1.7320508075688772f;   // sqrt(3)
  const float s5 = 2.23606797749979f;     // sqrt(5)
  const float s15 = 3.872983346207417f;   // sqrt(15)
  float* Ye = Y + (size_t)e * L2;
  Ye[0] = 1.0f;
  Ye[1] = s3 * x;  Ye[2] = s3 * y;  Ye[3] = s3 * z;
  Ye[4] = s15 * x * y;
  Ye[5] = s15 * y * z;
  Ye[6] = 0.5f * s5 * (3.0f * z * z - 1.0f);
  Ye[7] = s15 * x * z;
  Ye[8] = 0.5f * s15 * (x * x - y * y);
  float u  = r / RMAX;
  float u3 = u * u * u, u6 = u3 * u3, u7 = u6 * u, u8 = u7 * u;
  float fc = (u < 1.0f) ? (1.0f - 28.0f * u6 + 48.0f * u7 - 21.0f * u8) : 0.0f;
  float pref = sqrtf(2.0f / RMAX) * inv * fc;
  float w = PI_F * r / RMAX;
  float* efe = ef + (size_t)e * NB;
#pragma unroll
  for (int k = 0; k < NB; ++k) efe[k] = pref * sinf((float)(k + 1) * w);
}

// ------------------------------------- h0 = node_attrs @ W_embed, E0 atomic
__global__ void k_embed(const float* __restrict__ attrs,
                        const float* __restrict__ Wemb,
                        const float* __restrict__ aE,
                        const int* __restrict__ batch,
                        float* __restrict__ h, float* __restrict__ contrib) {
  int t = blockIdx.x * blockDim.x + threadIdx.x;
  if (t >= N * C) return;
  int n = t >> 6, c = t & 63;
  const float* a = attrs + (size_t)n * NE;
  float v = 0.0f;
#pragma unroll
  for (int k = 0; k < NE; ++k) v += a[k] * Wemb[k * C + c];
  h[(size_t)n * NL + c] = v;  // l=0 slot; l>0 pre-zeroed
  if (c == 0) {
    float e0 = 0.0f;
#pragma unroll
    for (int k = 0; k < NE; ++k) e0 += a[k] * aE[k];
    atomicAdd(&contrib[batch[n] * 3 + 0], e0);
  }
}

// ------------- Out[16 rows x 64] = A[16 x 64] @ W[64 x 64], one wave / tile
__global__ __launch_bounds__(32)
void k_gemm64(const float* __restrict__ A, const float* __restrict__ W,
              float* __restrict__ Out) {
  const int lane = threadIdx.x;
  const int half = lane >> 4;
  const int l15  = lane & 15;
  const size_t row0 = (size_t)blockIdx.x * 16;
  const float* Ar = A + (row0 + l15) * C;
  v8f acc0 = {0.f,0.f,0.f,0.f,0.f,0.f,0.f,0.f};
  v8f acc1 = acc0, acc2 = acc0, acc3 = acc0;
  for (int k0 = 0; k0 < C; k0 += 4) {
    v2f a = { Ar[k0 + 2 * half], Ar[k0 + 2 * half + 1] };
    const float* Wk = W + (size_t)(k0 + 2 * half) * C + l15;
    v2f b0 = { Wk[0],  Wk[C] };
    v2f b1 = { Wk[16], Wk[C + 16] };
    v2f b2 = { Wk[32], Wk[C + 32] };
    v2f b3 = { Wk[48], Wk[C + 48] };
    acc0 = wmma4(a, b0, acc0);
    acc1 = wmma4(a, b1, acc1);
    acc2 = wmma4(a, b2, acc2);
    acc3 = wmma4(a, b3, acc3);
  }
  float* Or = Out + row0 * C + l15;
#pragma unroll
  for (int v = 0; v < 8; ++v) {
    float* p = Or + (size_t)(v + 8 * half) * C;
    p[0]  = acc0[v];
    p[16] = acc1[v];
    p[32] = acc2[v];
    p[48] = acc3[v];
  }
}

// ---------- Out[16 x 64] = A[16 x 64] @ W^T  (Out[row,n] = sum_k A[row,k]*W[n,k])
__global__ __launch_bounds__(32)
void k_gemmT64(const float* __restrict__ A, const float* __restrict__ W,
               float* __restrict__ Out) {
  const int lane = threadIdx.x;
  const int half = lane >> 4;
  const int l15  = lane & 15;
  const size_t row0 = (size_t)blockIdx.x * 16;
  const float* Ar = A + (row0 + l15) * C;
  v8f acc0 = {0.f,0.f,0.f,0.f,0.f,0.f,0.f,0.f};
  v8f acc1 = acc0, acc2 = acc0, acc3 = acc0;
  for (int k0 = 0; k0 < C; k0 += 4) {
    int kk = k0 + 2 * half;
    v2f a = { Ar[kk], Ar[kk + 1] };
    const float* W0 = W + (size_t)l15 * C + kk;
    v2f b0 = { W0[0],       W0[1] };
    v2f b1 = { W0[16 * C],  W0[16 * C + 1] };
    v2f b2 = { W0[32 * C],  W0[32 * C + 1] };
    v2f b3 = { W0[48 * C],  W0[48 * C + 1] };
    acc0 = wmma4(a, b0, acc0);
    acc1 = wmma4(a, b1, acc1);
    acc2 = wmma4(a, b2, acc2);
    acc3 = wmma4(a, b3, acc3);
  }
  float* Or = Out + row0 * C + l15;
#pragma unroll
  for (int v = 0; v < 8; ++v) {
    float* p = Or + (size_t)(v + 8 * half) * C;
    p[0]  = acc0[v];
    p[16] = acc1[v];
    p[32] = acc2[v];
    p[48] = acc3[v];
  }
}

// ------- R[16 edges x 64] = silu(ef[16x8] @ W1[8x64]) @ W2[64x64], per wave
__global__ __launch_bounds__(32)
void k_radialR(const float* __restrict__ ef, const float* __restrict__ W1,
               const float* __restrict__ W2, float* __restrict__ Rout) {
  __shared__ float hid[16 * C];
  const int lane = threadIdx.x;
  const int half = lane >> 4;
  const int l15  = lane & 15;
  const size_t row0 = (size_t)blockIdx.x * 16;
  v8f h0 = {0.f,0.f,0.f,0.f,0.f,0.f,0.f,0.f};
  v8f h1 = h0, h2 = h0, h3 = h0;
  const float* er = ef + (row0 + l15) * NB;
  for (int k0 = 0; k0 < NB; k0 += 4) {
    v2f a = { er[k0 + 2 * half], er[k0 + 2 * half + 1] };
    const float* Wk = W1 + (size_t)(k0 + 2 * half) * C + l15;
    v2f b0 = { Wk[0],  Wk[C] };
    v2f b1 = { Wk[16], Wk[C + 16] };
    v2f b2 = { Wk[32], Wk[C + 32] };
    v2f b3 = { Wk[48], Wk[C + 48] };
    h0 = wmma4(a, b0, h0);
    h1 = wmma4(a, b1, h1);
    h2 = wmma4(a, b2, h2);
    h3 = wmma4(a, b3, h3);
  }
#pragma unroll
  for (int v = 0; v < 8; ++v) {
    float* p = hid + (v + 8 * half) * C + l15;
    p[0]  = siluf(h0[v]);
    p[16] = siluf(h1[v]);
    p[32] = siluf(h2[v]);
    p[48] = siluf(h3[v]);
  }
  __syncthreads();
  v8f r0 = {0.f,0.f,0.f,0.f,0.f,0.f,0.f,0.f};
  v8f r1 = r0, r2 = r0, r3 = r0;
  const float* Hr = hid + l15 * C;
  for (int k0 = 0; k0 < C; k0 += 4) {
    v2f a = { Hr[k0 + 2 * half], Hr[k0 + 2 * half + 1] };
    const float* Wk = W2 + (size_t)(k0 + 2 * half) * C + l15;
    v2f b0 = { Wk[0],  Wk[C] };
    v2f b1 = { Wk[16], Wk[C + 16] };
    v2f b2 = { Wk[32], Wk[C + 32] };
    v2f b3 = { Wk[48], Wk[C + 48] };
    r0 = wmma4(a, b0, r0);
    r1 = wmma4(a, b1, r1);
    r2 = wmma4(a, b2, r2);
    r3 = wmma4(a, b3, r3);
  }
  float* Or = Rout + row0 * C + l15;
#pragma unroll
  for (int v = 0; v < 8; ++v) {
    float* p = Or + (size_t)(v + 8 * half) * C;
    p[0]  = r0[v];
    p[16] = r1[v];
    p[32] = r2[v];
    p[48] = r3[v];
  }
}

// ---------- forward message: s = sum_l hu[snd,l,:]*Y[l]; scatter Y[l]*(R*s)
__global__ __launch_bounds__(256)
void k_msg(const int* __restrict__ snd, const int* __restrict__ rcv,
           const float* __restrict__ Y, const float* __restrict__ hu,
           const float* __restrict__ R, float* __restrict__ agg) {
  int t = blockIdx.x * blockDim.x + threadIdx.x;
  int e = t >> 6, c = t & 63;
  if (e >= E) return;
  int s = snd[e], d = rcv[e];
  const float* Ye  = Y + (size_t)e * L2;
  const float* hus = hu + (size_t)s * NL + c;
  float y[L2];
  float acc = 0.0f;
#pragma unroll
  for (int l = 0; l < L2; ++l) {
    y[l] = Ye[l];
    acc += hus[l * C] * y[l];
  }
  float tv = R[(size_t)e * C + c] * acc;
  float* ag = agg + (size_t)d * NL + c;
#pragma unroll
  for (int l = 0; l < L2; ++l) atomicAdd(ag + l * C, y[l] * tv);
}

// ---------- backward message: given dagg at rcv, produce dR, dhu (opt), dY
__global__ __launch_bounds__(256)
void k_msg_bwd(const int* __restrict__ snd, const int* __restrict__ rcv,
               const float* __restrict__ Y, const float* __restrict__ hu,
               const float* __restrict__ R, const float* __restrict__ dagg,
               float* __restrict__ dhu, float* __restrict__ dR,
               float* __restrict__ dY, int do_dhu) {
  __shared__ float red[4][L2][C];            // 9 KB
  int t = blockIdx.x * blockDim.x + threadIdx.x;   // exact grid: E*C threads
  int e = t >> 6, c = t & 63;
  int sub = threadIdx.x >> 6;
  int s = snd[e], d = rcv[e];
  const float* Ye  = Y + (size_t)e * L2;
  const float* hus = hu + (size_t)s * NL + c;
  const float* dag = dagg + (size_t)d * NL + c;
  float y[L2], hv[L2], da[L2];
  float sv = 0.0f, dt = 0.0f;
#pragma unroll
  for (int l = 0; l < L2; ++l) {
    y[l]  = Ye[l];
    hv[l] = hus[l * C];
    da[l] = dag[l * C];
    sv += hv[l] * y[l];
    dt += da[l] * y[l];
  }
  float Rv   = R[(size_t)e * C + c];
  float tval = Rv * sv;          // forward t
  float ds   = dt * Rv;
  dR[(size_t)e * C + c] = dt * sv;
  if (do_dhu) {
    float* dh = dhu + (size_t)s * NL + c;
#pragma unroll
    for (int l = 0; l < L2; ++l) atomicAdd(dh + l * C, ds * y[l]);
  }
#pragma unroll
  for (int l = 0; l < L2; ++l) red[sub][l][c] = da[l] * tval + ds * hv[l];
  __syncthreads();
  if (c < L2) {
    const float* rr = red[sub][c];
    float acc = 0.0f;
    for (int cc = 0; cc < C; ++cc) acc += rr[cc];
    dY[(size_t)e * L2 + c] += acc;     // single writer per element
  }
}

// ------- radial backward: dH = dR@W2^T; P = ef@W1; dP = dH*silu'(P); def += dP@W1^T
__global__ __launch_bounds__(32)
void k_radial_bwd(const float* __restrict__ ef, const float* __restrict__ W1,
                  const float* __restrict__ W2, const float* __restrict__ dR,
                  float* __restrict__ def) {
  __shared__ float dp[16 * C];
  const int lane = threadIdx.x;
  const int half = lane >> 4;
  const int l15  = lane & 15;
  const size_t row0 = (size_t)blockIdx.x * 16;
  // dH = dR @ W2^T
  v8f a0 = {0.f,0.f,0.f,0.f,0.f,0.f,0.f,0.f};
  v8f a1 = a0, a2 = a0, a3 = a0;
  const float* dRr = dR + (row0 + l15) * C;
  for (int k0 = 0; k0 < C; k0 += 4) {
    int kk = k0 + 2 * half;
    v2f a = { dRr[kk], dRr[kk + 1] };
    const float* W0 = W2 + (size_t)l15 * C + kk;
    v2f b0 = { W0[0],      W0[1] };
    v2f b1 = { W0[16 * C], W0[16 * C + 1] };
    v2f b2 = { W0[32 * C], W0[32 * C + 1] };
    v2f b3 = { W0[48 * C], W0[48 * C + 1] };
    a0 = wmma4(a, b0, a0);
    a1 = wmma4(a, b1, a1);
    a2 = wmma4(a, b2, a2);
    a3 = wmma4(a, b3, a3);
  }
  // recompute P = ef @ W1
  v8f p0 = {0.f,0.f,0.f,0.f,0.f,0.f,0.f,0.f};
  v8f p1 = p0, p2 = p0, p3 = p0;
  const float* er = ef + (row0 + l15) * NB;
  for (int k0 = 0; k0 < NB; k0 += 4) {
    v2f a = { er[k0 + 2 * half], er[k0 + 2 * half + 1] };
    const float* Wk = W1 + (size_t)(k0 + 2 * half) * C + l15;
    v2f b0 = { Wk[0],  Wk[C] };
    v2f b1 = { Wk[16], Wk[C + 16] };
    v2f b2 = { Wk[32], Wk[C + 32] };
    v2f b3 = { Wk[48], Wk[C + 48] };
    p0 = wmma4(a, b0, p0);
    p1 = wmma4(a, b1, p1);
    p2 = wmma4(a, b2, p2);
    p3 = wmma4(a, b3, p3);
  }
  // dP = dH * silu'(P) -> row-major LDS
#pragma unroll
  for (int v = 0; v < 8; ++v) {
    float* q = dp + (v + 8 * half) * C + l15;
    q[0]  = a0[v] * dsiluf(p0[v]);
    q[16] = a1[v] * dsiluf(p1[v]);
    q[32] = a2[v] * dsiluf(p2[v]);
    q[48] = a3[v] * dsiluf(p3[v]);
  }
  __syncthreads();
  // def = dP @ W1^T  (output 16x8; B zero-padded to 16 cols)
  v8f d0 = {0.f,0.f,0.f,0.f,0.f,0.f,0.f,0.f};
  const float* Pr = dp + l15 * C;
  for (int k0 = 0; k0 < C; k0 += 4) {
    int kk = k0 + 2 * half;
    v2f a = { Pr[kk], Pr[kk + 1] };
    v2f b = { 0.f, 0.f };
    if (l15 < NB) {
      const float* Wc = W1 + (size_t)l15 * C + kk;   // W1^T[k][col] = W1[col][k]
      b.x = Wc[0];
      b.y = Wc[1];
    }
    d0 = wmma4(a, b, d0);
  }
  if (l15 < NB) {
#pragma unroll
    for (int v = 0; v < 8; ++v) {
      def[(row0 + v + 8 * half) * NB + l15] += d0[v];   // single writer
    }
  }
}

// ---------------------------------------- h[:,0,:] += node_attrs @ W_skip
__global__ void k_skip(float* __restrict__ h, const float* __restrict__ attrs,
                       const float* __restrict__ Wskip) {
  int t = blockIdx.x * blockDim.x + threadIdx.x;
  if (t >= N * C) return;
  int n = t >> 6, c = t & 63;
  const float* a = attrs + (size_t)n * NE;
  float v = 0.0f;
#pragma unroll
  for (int k = 0; k < NE; ++k) v += a[k] * Wskip[k * C + c];
  h[(size_t)n * NL + c] += v;
}

// ---------------------------------------------------- readouts -> contrib
__global__ void k_read0(const float* __restrict__ h, const float* __restrict__ w0,
                        const int* __restrict__ batch, float* __restrict__ contrib) {
  int n = blockIdx.x * blockDim.x + threadIdx.x;
  if (n >= N) return;
  const float* hn = h + (size_t)n * NL;
  float en = 0.0f;
  for (int c = 0; c < C; ++c) en += hn[c] * w0[c];
  atomicAdd(&contrib[batch[n] * 3 + 1], en);
}

__global__ void k_read1(const float* __restrict__ h, const float* __restrict__ Wr1,
                        const float* __restrict__ wr2, const int* __restrict__ batch,
                        float* __restrict__ contrib) {
  int n = blockIdx.x * blockDim.x + threadIdx.x;
  if (n >= N) return;
  const float* hn = h + (size_t)n * NL;
  float en = 0.0f;
  for (int j = 0; j < HID; ++j) {
    float p = 0.0f;
    for (int c = 0; c < C; ++c) p += hn[c] * Wr1[c * HID + j];
    en += siluf(p) * wr2[j];
  }
  atomicAdd(&contrib[batch[n] * 3 + 2], en);
}

// ------------------------- readout backward: g[:,0,:] = d(en)/dh  (layer 2)
__global__ void k_read1_bwd(const float* __restrict__ h, const float* __restrict__ Wr1,
                            const float* __restrict__ wr2, float* __restrict__ g) {
  int n = blockIdx.x * blockDim.x + threadIdx.x;
  if (n >= N) return;
  const float* hn = h + (size_t)n * NL;
  float a[HID];
  for (int j = 0; j < HID; ++j) {
    float p = 0.0f;
    for (int c = 0; c < C; ++c) p += hn[c] * Wr1[c * HID + j];
    a[j] = dsiluf(p) * wr2[j];
  }
  float* gn = g + (size_t)n * NL;
  for (int c = 0; c < C; ++c) {
    float v = 0.0f;
#pragma unroll
    for (int j = 0; j < HID; ++j) v += a[j] * Wr1[c * HID + j];
    gn[c] = v;                       // l=0 slot; l>0 pre-zeroed
  }
}

// ---------------------- readout-0 backward: g[:,0,:] += w_read0  (layer 1)
__global__ void k_read0_grad(float* __restrict__ g, const float* __restrict__ w0) {
  int t = blockIdx.x * blockDim.x + threadIdx.x;
  if (t >= N * C) return;
  int n = t >> 6, c = t & 63;
  g[(size_t)n * NL + c] += w0[c];
}

// ------------------ final: (dY, def) -> dvec -> force atomics on snd / rcv
__global__ void k_force(const int* __restrict__ snd, const int* __restrict__ rcv,
                        const float* __restrict__ rbuf, const float* __restrict__ ubuf,
                        const float* __restrict__ dY, const float* __restrict__ defb,
                        float* __restrict__ forces) {
  int e = blockIdx.x * blockDim.x + threadIdx.x;
  if (e >= E) return;
  float r = rbuf[e];
  float inv = 1.0f / r;
  float x = ubuf[3 * e], y = ubuf[3 * e + 1], z = ubuf[3 * e + 2];
  const float s3 = 1.7320508075688772f;
  const float s5 = 2.23606797749979f;
  const float s15 = 3.872983346207417f;
  const float* dYe = dY + (size_t)e * L2;
  // g_u = sum_l dY_l * grad_u Y_l
  float gx = s3 * dYe[1] + s15 * y * dYe[4] + s15 * z * dYe[7] + s15 * x * dYe[8];
  float gy = s3 * dYe[2] + s15 * x * dYe[4] + s15 * z * dYe[5] - s15 * y * dYe[8];
  float gz = s3 * dYe[3] + s15 * y * dYe[5] + 3.0f * s5 * z * dYe[6] + s15 * x * dYe[7];
  float gdotu = gx * x + gy * y + gz * z;
  // dr from def: ef_k = A0 * sin(k*pi*r/RMAX) * fc(r/RMAX) / r
  float u  = r / RMAX;
  float u2 = u * u, u3 = u2 * u, u5 = u3 * u2, u6 = u5 * u, u7 = u6 * u, u8 = u7 * u;
  float fc, dfc;
  if (u < 1.0f) {
    fc  = 1.0f - 28.0f * u6 + 48.0f * u7 - 21.0f * u8;
    dfc = (-168.0f * u5 + 336.0f * u6 - 168.0f * u7) / RMAX;
  } else {
    fc = 0.0f; dfc = 0.0f;
  }
  float A0 = sqrtf(2.0f / RMAX);
  float w = PI_F * r / RMAX;
  float drt = 0.0f;
  const float* de = defb + (size_t)e * NB;
#pragma unroll
  for (int k = 1; k <= NB; ++k) {
    float sk = sinf((float)k * w);
    float ck = cosf((float)k * w);
    float dek = A0 * (((float)k * PI_F / RMAX) * ck * fc * inv
                      - sk * fc * inv * inv + sk * dfc * inv);
    drt += de[k - 1] * dek;
  }
  // dvec = (g_u - (g_u.u)u)/r + dr*u ;  force[snd] -= dvec, force[rcv] += dvec
  float dvx = (gx - gdotu * x) * inv + drt * x;
  float dvy = (gy - gdotu * y) * inv + drt * y;
  float dvz = (gz - gdotu * z) * inv + drt * z;
  int s = snd[e], d = rcv[e];
  atomicAdd(&forces[3 * s],     -dvx);
  atomicAdd(&forces[3 * s + 1], -dvy);
  atomicAdd(&forces[3 * s + 2], -dvz);
  atomicAdd(&forces[3 * d],      dvx);
  atomicAdd(&forces[3 * d + 1],  dvy);
  atomicAdd(&forces[3 * d + 2],  dvz);
}

// ------------------------------------------------ total = sum contributions
__global__ void k_total(const float* __restrict__ contrib, float* __restrict__ total) {
  int g = threadIdx.x;
  if (g < G) total[g] = contrib[g * 3] + contrib[g * 3 + 1] + contrib[g * 3 + 2];
}

}  // namespace

extern "C" void kernel_launch(void* const* d_in, const int* in_sizes, int n_in,
                              void* d_out, int out_size, void* d_ws, size_t ws_size,
                              hipStream_t stream) {
  const float* positions       = (const float*)d_in[0];
  const float* node_attrs      = (const float*)d_in[1];
  const float* shifts          = (const float*)d_in[2];
  const float* atomic_energies = (const float*)d_in[3];
  const float* W_embed         = (const float*)d_in[4];
  const float* W_up            = (const float*)d_in[5];
  const float* W1              = (const float*)d_in[6];
  const float* W2              = (const float*)d_in[7];
  const float* W_out           = (const float*)d_in[8];
  const float* W_skip          = (const float*)d_in[9];
  const float* w_read0         = (const float*)d_in[10];
  const float* W_r1            = (const float*)d_in[11];
  const float* w_r2            = (const float*)d_in[12];
  const int*   edge_index      = (const int*)d_in[13];
  const int*   batch           = (const int*)d_in[14];
  const int* snd = edge_index;
  const int* rcv = edge_index + E;

  float* out     = (float*)d_out;
  float* total   = out;              // [16]
  float* contrib = out + G;          // [16*3]
  float* forces  = out + G + G * 3;  // [N*3]

  float* ws = (float*)d_ws;
  size_t o = 0;
  float* Y    = ws + o; o += (size_t)E * L2;
  float* ef   = ws + o; o += (size_t)E * NB;
  float* rbuf = ws + o; o += (size_t)E;
  float* ubuf = ws + o; o += (size_t)E * 3;
  float* hu1  = ws + o; o += (size_t)N * NL;
  float* hu2  = ws + o; o += (size_t)N * NL;
  float* R1   = ws + o; o += (size_t)E * C;
  float* R2   = ws + o; o += (size_t)E * C;
  float* dRb  = ws + o; o += (size_t)E * C;
  float* agg  = ws + o; o += (size_t)N * NL;   // also dagg in backward
  float* hA   = ws + o; o += (size_t)N * NL;
  float* hB   = ws + o; o += (size_t)N * NL;
  float* dhu  = ws + o; o += (size_t)N * NL;
  float* g    = ws + o; o += (size_t)N * NL;
  float* dY   = ws + o; o += (size_t)E * L2;
  float* defb = ws + o; o += (size_t)E * NB;

  float* hu_l[2] = { hu1, hu2 };
  float* R_l[2]  = { R1, R2 };

  auto zero = [&](float* p, long n) {
    k_zero<<<(int)((n + 255) / 256), 256, 0, stream>>>(p, n);
  };

  // ---------------- forward ----------------
  zero(out, (long)out_size);
  zero(hA, (long)N * NL);
  k_geom<<<(E + 255) / 256, 256, 0, stream>>>(positions, shifts, snd, rcv,
                                              Y, ef, rbuf, ubuf);
  k_embed<<<(N * C + 255) / 256, 256, 0, stream>>>(node_attrs, W_embed,
                                                   atomic_energies, batch, hA, contrib);

  float* hc = hA;
  float* hn = hB;
  for (int i = 0; i < 2; ++i) {
    zero(agg, (long)N * NL);
    k_gemm64<<<(N * L2) / 16, 32, 0, stream>>>(hc, W_up + (size_t)i * C * C, hu_l[i]);
    k_radialR<<<E / 16, 32, 0, stream>>>(ef, W1 + (size_t)i * NB * C,
                                         W2 + (size_t)i * C * C, R_l[i]);
    k_msg<<<(E * C) / 256, 256, 0, stream>>>(snd, rcv, Y, hu_l[i], R_l[i], agg);
    k_gemm64<<<(N * L2) / 16, 32, 0, stream>>>(agg, W_out + (size_t)i * C * C, hn);
    k_skip<<<(N * C + 255) / 256, 256, 0, stream>>>(hn, node_attrs,
                                                    W_skip + (size_t)i * NE * C);
    if (i == 0)
      k_read0<<<(N + 255) / 256, 256, 0, stream>>>(hn, w_read0, batch, contrib);
    else
      k_read1<<<(N + 255) / 256, 256, 0, stream>>>(hn, W_r1, w_r2, batch, contrib);
    float* tmp = hc; hc = hn; hn = tmp;
  }
  k_total<<<1, 32, 0, stream>>>(contrib, total);
  float* h2p = hc;   // h after layer 2 (readout-1 input)

  // ---------------- backward (forces) ----------------
  zero(dY, (long)E * L2);
  zero(defb, (long)E * NB);

  // layer 2 (i = 1)
  zero(g, (long)N * NL);
  k_read1_bwd<<<(N + 255) / 256, 256, 0, stream>>>(h2p, W_r1, w_r2, g);
  k_gemmT64<<<(N * L2) / 16, 32, 0, stream>>>(g, W_out + (size_t)1 * C * C, agg);
  zero(dhu, (long)N * NL);
  k_msg_bwd<<<(E * C) / 256, 256, 0, stream>>>(snd, rcv, Y, hu2, R2, agg,
                                               dhu, dRb, dY, 1);
  k_radial_bwd<<<E / 16, 32, 0, stream>>>(ef, W1 + (size_t)1 * NB * C,
                                          W2 + (size_t)1 * C * C, dRb, defb);
  k_gemmT64<<<(N * L2) / 16, 32, 0, stream>>>(dhu, W_up + (size_t)1 * C * C, g);

  // layer 1 (i = 0):  g currently = dE/dh1'
  k_read0_grad<<<(N * C + 255) / 256, 256, 0, stream>>>(g, w_read0);
  k_gemmT64<<<(N * L2) / 16, 32, 0, stream>>>(g, W_out, agg);
  k_msg_bwd<<<(E * C) / 256, 256, 0, stream>>>(snd, rcv, Y, hu1, R1, agg,
                                               dhu, dRb, dY, 0);  // h0 is pos-independent
  k_radial_bwd<<<E / 16, 32, 0, stream>>>(ef, W1, W2, dRb, defb);

  k_force<<<(E + 255) / 256, 256, 0, stream>>>(snd, rcv, rbuf, ubuf, dY, defb, forces);
}